// Retriever_38972533244620
// MI455X (gfx1250) — compile-verified
//
#include <hip/hip_runtime.h>

typedef __attribute__((ext_vector_type(2))) float v2f;
typedef __attribute__((ext_vector_type(8))) float v8f;

#define GROUPS 8
#define POOL   64
#define CH     96
#define BATCH  4
#define TOPK   4
#define KDIM   (GROUPS * CH)   /* 768 flattened contraction length */

// ---------------------------------------------------------------------------
// Kernel A: cosine-sim scores via V_WMMA_F32_16X16X4_F32, mean, top-4, weights
// ---------------------------------------------------------------------------
__global__ __launch_bounds__(256) void retr_score_kernel(
    const float* __restrict__ q,        // [4, 768]
    const float* __restrict__ keys,     // [1, 8, 64, 96] -> row r = g*64+p
    int*   __restrict__ ws_idx,         // [4] selected pool indices
    float* __restrict__ ws_w,           // [4] normalized weights
    float* __restrict__ out_tail)       // [8] = idx_vote(as float) ++ dis_weight
{
    __shared__ float qn_l[BATCH * GROUPS * CH];   // 12 KB
    __shared__ float s_l[KDIM];                   // sum over batch of qn
    __shared__ float kinv[GROUPS * POOL];         // per-key-row inverse norms
    __shared__ float sim_l[POOL];

    const int tid = threadIdx.x;

    // ---- normalize query rows (32 rows of 96; row id = b*8+g) ----
    if (tid < BATCH * GROUPS) {
        const float* row = q + tid * CH;
        float ss = 0.f;
        #pragma unroll 4
        for (int c = 0; c < CH; ++c) { float v = row[c]; ss = fmaf(v, v, ss); }
        const float inv = 1.f / fmaxf(sqrtf(ss), 1e-8f);
        for (int c = 0; c < CH; ++c) qn_l[tid * CH + c] = row[c] * inv;
    }
    // ---- key row inverse norms (512 rows of 96) ----
    for (int r = tid; r < GROUPS * POOL; r += 256) {
        const float* row = keys + r * CH;
        float ss = 0.f;
        #pragma unroll 4
        for (int c = 0; c < CH; ++c) { float v = row[c]; ss = fmaf(v, v, ss); }
        kinv[r] = 1.f / fmaxf(sqrtf(ss), 1e-8f);
    }
    __syncthreads();

    // ---- s[g*96+c] = sum_b qn[b,g,c]  (fold the batch+mean into the GEMM) ----
    for (int i = tid; i < KDIM; i += 256) {
        const int g = i / CH, c = i - g * CH;
        float acc = 0.f;
        #pragma unroll
        for (int b = 0; b < BATCH; ++b) acc += qn_l[(b * GROUPS + g) * CH + c];
        s_l[i] = acc;
    }
    __syncthreads();

    // ---- wave 0: idx_sim[p] = (1/32) * sum_k kn[p,k] * s[k] via WMMA f32 ----
    // A (16x4): lane m = pool row; lanes 0-15 hold K={0,1}, lanes 16-31 K={2,3}.
    // B (4x16): s[k] broadcast across all N columns (lane->N mapping irrelevant).
    // D (16x16): every column identical; lane 0 holds rows 0-7, lane 16 rows 8-15.
    if (tid < 32) {
        const int lane = tid;
        const int m  = lane & 15;
        const int hi = lane >> 4;          // 0: K 0,1 / rows 0-7;  1: K 2,3 / rows 8-15
        v8f acc[4] = {};                   // 4 chains of 16 pools each
        for (int k = 0; k < KDIM; k += 4) {       // 96 % 4 == 0: never crosses a group
            const int g  = k / CH;
            const int c0 = (k - g * CH) + 2 * hi;
            v2f bv;
            bv.x = s_l[k + 2 * hi];
            bv.y = s_l[k + 2 * hi + 1];
            #pragma unroll
            for (int t = 0; t < 4; ++t) {
                const int r = g * POOL + t * 16 + m;
                const float kv = kinv[r];
                v2f av;
                av.x = keys[r * CH + c0]     * kv;
                av.y = keys[r * CH + c0 + 1] * kv;
                acc[t] = __builtin_amdgcn_wmma_f32_16x16x4_f32(
                    false, av, false, bv, (short)0, acc[t], false, false);
            }
        }
        if (m == 0) {      // lane 0 -> rows 0-7 of each chain, lane 16 -> rows 8-15
            #pragma unroll
            for (int t = 0; t < 4; ++t)
                #pragma unroll
                for (int j = 0; j < 8; ++j)
                    sim_l[t * 16 + hi * 8 + j] = acc[t][j] * (1.0f / 32.0f);
        }
    }
    __syncthreads();

    // ---- serial top-4 (matches lax.top_k first-occurrence tie-break) ----
    if (tid == 0) {
        unsigned long long used = 0ull;
        int   idxs[TOPK];
        float vals[TOPK];
        float wsum = 0.f;
        for (int kk = 0; kk < TOPK; ++kk) {
            float best = -3.402823e38f; int bi = 0;
            for (int p = 0; p < POOL; ++p)
                if (!((used >> p) & 1ull) && sim_l[p] > best) { best = sim_l[p]; bi = p; }
            used |= (1ull << bi);
            idxs[kk] = bi; vals[kk] = best; wsum += best;
        }
        const float inv = 1.f / (wsum + 1e-9f);
        #pragma unroll
        for (int kk = 0; kk < TOPK; ++kk) {
            ws_idx[kk] = idxs[kk];
            ws_w[kk]   = vals[kk] * inv;
            out_tail[kk]        = (float)idxs[kk];   // idx_vote (output 2)
            out_tail[TOPK + kk] = vals[kk] * inv;    // dis_weight (output 3)
        }
    }
}

// ---------------------------------------------------------------------------
// Kernel B: combined = sum_k w[k]*woc[idx[k]]; broadcast to 4 batch copies.
// Pure streaming: 4x b128 loads + 4 FMAs + 4x b128 stores per thread.
// ---------------------------------------------------------------------------
__global__ __launch_bounds__(256) void retr_combine_kernel(
    const float* __restrict__ woc,   // [64, 2, L] -> 2*L floats per pool entry
    const int*   __restrict__ ws_idx,
    const float* __restrict__ ws_w,
    float*       __restrict__ out,   // [4, 2, L]
    long tl)                          // tl = 2*L
{
    const long n4 = tl >> 2;                                  // float4 count
    const long i4 = (long)blockIdx.x * blockDim.x + threadIdx.x;
    if (i4 >= n4) return;

    const int   i0 = ws_idx[0], i1 = ws_idx[1], i2 = ws_idx[2], i3 = ws_idx[3];
    const float w0 = ws_w[0],   w1 = ws_w[1],   w2 = ws_w[2],   w3 = ws_w[3];

    const float4* p0 = (const float4*)(woc + (size_t)i0 * tl);
    const float4* p1 = (const float4*)(woc + (size_t)i1 * tl);
    const float4* p2 = (const float4*)(woc + (size_t)i2 * tl);
    const float4* p3 = (const float4*)(woc + (size_t)i3 * tl);

    const float4 a = p0[i4];
    const float4 b = p1[i4];
    const float4 c = p2[i4];
    const float4 d = p3[i4];

    float4 r;
    r.x = fmaf(w3, d.x, fmaf(w2, c.x, fmaf(w1, b.x, w0 * a.x)));
    r.y = fmaf(w3, d.y, fmaf(w2, c.y, fmaf(w1, b.y, w0 * a.y)));
    r.z = fmaf(w3, d.z, fmaf(w2, c.z, fmaf(w1, b.z, w0 * a.z)));
    r.w = fmaf(w3, d.w, fmaf(w2, c.w, fmaf(w1, b.w, w0 * a.w)));

    float4* o = (float4*)out;
    o[i4]          = r;   // batch 0
    o[i4 + n4]     = r;   // batch 1
    o[i4 + 2 * n4] = r;   // batch 2
    o[i4 + 3 * n4] = r;   // batch 3
}

// ---------------------------------------------------------------------------
extern "C" void kernel_launch(void* const* d_in, const int* in_sizes, int n_in,
                              void* d_out, int out_size, void* d_ws, size_t ws_size,
                              hipStream_t stream) {
    const float* q    = (const float*)d_in[0];   // [4, 768]
    const float* keys = (const float*)d_in[1];   // [1, 8, 64, 96]
    const float* woc  = (const float*)d_in[2];   // [64, 2, L]
    float* out = (float*)d_out;

    int*   ws_idx = (int*)d_ws;
    float* ws_w   = (float*)d_ws + TOPK;

    const long L  = (long)in_sizes[2] / (POOL * 2);   // 524288
    const long tl = 2 * L;                            // floats per pool entry
    float* out_tail = out + (long)BATCH * tl;         // after updated_lora

    retr_score_kernel<<<1, 256, 0, stream>>>(q, keys, ws_idx, ws_w, out_tail);

    const long n4 = tl >> 2;                          // 262144 float4 elements
    const int  blocks = (int)((n4 + 255) / 256);      // 1024
    retr_combine_kernel<<<blocks, 256, 0, stream>>>(woc, ws_idx, ws_w, out, tl);
}